// GNNEncoder_19645180412415
// MI455X (gfx1250) — compile-verified
//
#include <hip/hip_runtime.h>
#include <hip/hip_bf16.h>

// ---------------------------------------------------------------------------
// GCN encoder, MI455X (gfx1250, wave32, WMMA).
//   Z = X @ W^T + b            -> v_wmma_f32_16x16x4_f32 (exact fp32 GEMM)
//   H = D^-1/2 (A+I) D^-1/2 Z  -> bitmap-sparse aggregation (L2 resident)
//   out = BN(H) [,ReLU]        -> deterministic two-stage column stats
// ---------------------------------------------------------------------------

#define NNODES 8192
#define DFEAT  256
#define WORDS_PER_ROW (NNODES / 32)          // 256 u32 per bitmap row
#define BITMAP_WORDS  (NNODES * WORDS_PER_ROW)
#define NPART  64                            // BN stat partitions (128 rows each)

typedef float v2f __attribute__((ext_vector_type(2)));
typedef float v8f __attribute__((ext_vector_type(8)));

// ---------------- adjacency construction ----------------

__global__ void k_init_bitmap(unsigned* __restrict__ bitmap) {
    int w = blockIdx.x * blockDim.x + threadIdx.x;
    if (w >= BITMAP_WORDS) return;
    int row = w >> 8;            // w / WORDS_PER_ROW
    int cw  = w & 255;
    // self-loop bit on the diagonal word
    bitmap[w] = (cw == (row >> 5)) ? (1u << (row & 31)) : 0u;
}

__global__ void k_scatter_edges(const int* __restrict__ ei,
                                unsigned* __restrict__ bitmap, int E) {
    int e = blockIdx.x * blockDim.x + threadIdx.x;
    if (e >= E) return;
    int r = ei[e];
    int c = ei[E + e];
    atomicOr(&bitmap[(size_t)r * WORDS_PER_ROW + (c >> 5)], 1u << (c & 31));
}

__global__ void k_degree_inv(const unsigned* __restrict__ bitmap,
                             float* __restrict__ dinv) {
    int row = blockIdx.x * blockDim.x + threadIdx.x;
    if (row >= NNODES) return;
    const uint4* p = (const uint4*)(bitmap + (size_t)row * WORDS_PER_ROW);
    int deg = 0;
    #pragma unroll 4
    for (int w = 0; w < WORDS_PER_ROW / 4; ++w) {
        uint4 v = p[w];
        deg += __popc(v.x) + __popc(v.y) + __popc(v.z) + __popc(v.w);
    }
    dinv[row] = (deg > 0) ? rsqrtf((float)deg) : 0.0f;
}

// ---------------- dense linear layer: Z = X @ W^T + b (WMMA fp32) ----------

// One wave32 computes one 16x16 tile of Z. K stepped by 4.
// A (16x4 f32) per ISA: lanes 0-15 hold row M=lane with K={0,1} in v0,v1;
// lanes 16-31 hold the same rows with K={2,3}. B (4x16) mirrors with M->N.
// C/D: VGPR r, lane<16 -> (M=r, N=lane); lane>=16 -> (M=r+8, N=lane-16).
__global__ void k_linear_wmma(const float* __restrict__ X,
                              const float* __restrict__ W,   // [DFEAT x DFEAT] row-major (dout x din)
                              const float* __restrict__ bias,
                              float* __restrict__ Z) {
    const int wave = (blockIdx.x * blockDim.x + threadIdx.x) >> 5;
    const int lane = threadIdx.x & 31;
    const int tilesN = DFEAT / 16;                 // 16
    const int tm = wave / tilesN;                  // 0..511
    const int tn = wave % tilesN;                  // 0..15
    const int half = lane >> 4;                    // 0: K pair {0,1}, 1: {2,3}
    const int l    = lane & 15;
    const int rowA = tm * 16 + l;                  // output row handled by lane
    const int colB = tn * 16 + l;                  // output col handled by lane

    const float* __restrict__ xrow = X + (size_t)rowA * DFEAT;
    const float* __restrict__ wrow = W + (size_t)colB * DFEAT; // B(k,n) = W[n][k]

    v8f acc = {};
    #pragma unroll 8
    for (int k = 0; k < DFEAT; k += 4) {
        const int ka = k + 2 * half;
        v2f a, b;
        a.x = xrow[ka];
        a.y = xrow[ka + 1];
        b.x = wrow[ka];
        b.y = wrow[ka + 1];
        acc = __builtin_amdgcn_wmma_f32_16x16x4_f32(
            /*neg_a=*/false, a, /*neg_b=*/false, b,
            /*c_mod=*/(short)0, acc, /*reuse_a=*/false, /*reuse_b=*/false);
    }

    const int n = tn * 16 + l;
    const float bv = bias[n];
    #pragma unroll
    for (int r = 0; r < 8; ++r) {
        int m = tm * 16 + r + 8 * half;
        Z[(size_t)m * DFEAT + n] = acc[r] + bv;
    }
}

// ---------------- sparse normalized aggregation -----------------

__global__ void k_aggregate(const float* __restrict__ Z,
                            const float* __restrict__ dinv,
                            const unsigned* __restrict__ bitmap,
                            float* __restrict__ H) {
    const int row = blockIdx.x;          // one block per destination node
    const int d   = threadIdx.x;         // feature index, 256 threads
    const unsigned* __restrict__ wrow = bitmap + (size_t)row * WORDS_PER_ROW;

    float acc = 0.0f;
    for (int w = 0; w < WORDS_PER_ROW; ++w) {
        // stream the bitmap row; keep a few words ahead warm in cache
        __builtin_prefetch(wrow + w + 8, 0, 0);       // -> global_prefetch_b8
        unsigned bits = wrow[w];         // uniform across the block -> scalar load
        while (bits) {
            const int bit = __builtin_ctz(bits);
            bits &= bits - 1u;
            const int j = (w << 5) + bit;
            acc = fmaf(dinv[j], Z[(size_t)j * DFEAT + d], acc);
        }
    }
    H[(size_t)row * DFEAT + d] = acc * dinv[row];
}

// ---------------- batch-norm (deterministic fixed-order stats) ----------------

// Stage 1: 64 partitions x 128 contiguous rows, one thread per column.
// Coalesced reads; fixed accumulation order -> bit-deterministic.
__global__ void k_bn_stats(const float* __restrict__ H,
                           float* __restrict__ pSum,
                           float* __restrict__ pSq) {
    const int part = blockIdx.x;              // 0..NPART-1
    const int d    = threadIdx.x;             // column
    const float* __restrict__ p = H + (size_t)part * (NNODES / NPART) * DFEAT + d;
    float s = 0.0f, q = 0.0f;
    #pragma unroll 4
    for (int r = 0; r < NNODES / NPART; ++r) {
        const float v = p[(size_t)r * DFEAT];
        s += v;
        q = fmaf(v, v, q);
    }
    pSum[part * DFEAT + d] = s;
    pSq [part * DFEAT + d] = q;
}

// Stage 2: fold the 64 partials in fixed order, produce affine scale/shift.
__global__ void k_bn_finalize(const float* __restrict__ pSum,
                              const float* __restrict__ pSq,
                              const float* __restrict__ g,
                              const float* __restrict__ be,
                              float* __restrict__ scale,
                              float* __restrict__ shift) {
    const int d = threadIdx.x;
    float s = 0.0f, q = 0.0f;
    #pragma unroll 8
    for (int p = 0; p < NPART; ++p) {
        s += pSum[p * DFEAT + d];
        q += pSq [p * DFEAT + d];
    }
    const float inv_n = 1.0f / (float)NNODES;
    const float mu  = s * inv_n;
    const float var = q * inv_n - mu * mu;      // biased variance
    const float rs  = rsqrtf(var + 1e-5f);
    const float sc  = rs * g[d];
    scale[d] = sc;
    shift[d] = be[d] - mu * sc;
}

__global__ void k_bn_apply(const float* __restrict__ Hin,
                           const float* __restrict__ scale,
                           const float* __restrict__ shift,
                           float* __restrict__ out, int relu) {
    const int i = blockIdx.x * blockDim.x + threadIdx.x;   // NNODES*DFEAT threads
    const int d = i & (DFEAT - 1);
    float v = fmaf(Hin[i], scale[d], shift[d]);
    if (relu) v = fmaxf(v, 0.0f);
    out[i] = v;
}

// ---------------- workspace layout (bytes) ----------------
#define OFF_BITMAP 0u
#define OFF_DINV   (8u * 1024u * 1024u)                       // 8 MB bitmap
#define OFF_Z      (OFF_DINV + 32u * 1024u)                   // 32 KB dinv
#define OFF_H      (OFF_Z + 8u * 1024u * 1024u)               // 8 MB Z
#define OFF_STATS  (OFF_H + 8u * 1024u * 1024u)               // 8 MB H
// stats: pSum[64*256] | pSq[64*256] | scale[256] | shift[256]  (~132 KB)

extern "C" void kernel_launch(void* const* d_in, const int* in_sizes, int n_in,
                              void* d_out, int out_size, void* d_ws, size_t ws_size,
                              hipStream_t stream) {
    const float* x  = (const float*)d_in[0];
    const int*   ei = (const int*)d_in[1];
    const float* Wm[3] = { (const float*)d_in[2], (const float*)d_in[4], (const float*)d_in[6] };
    const float* bv[3] = { (const float*)d_in[3], (const float*)d_in[5], (const float*)d_in[7] };
    const float* gm[3] = { (const float*)d_in[8], (const float*)d_in[10], (const float*)d_in[12] };
    const float* bt[3] = { (const float*)d_in[9], (const float*)d_in[11], (const float*)d_in[13] };

    char* ws = (char*)d_ws;
    unsigned* bitmap = (unsigned*)(ws + OFF_BITMAP);
    float*    dinv   = (float*)(ws + OFF_DINV);
    float*    Z      = (float*)(ws + OFF_Z);
    float*    H      = (float*)(ws + OFF_H);
    float*    pSum   = (float*)(ws + OFF_STATS);
    float*    pSq    = pSum + NPART * DFEAT;
    float*    scale  = pSum + 2 * NPART * DFEAT;
    float*    shift  = scale + DFEAT;

    const int E = in_sizes[1] / 2;

    // ---- graph structure (built every call: deterministic, capture-safe) ----
    k_init_bitmap<<<BITMAP_WORDS / 256, 256, 0, stream>>>(bitmap);
    k_scatter_edges<<<(E + 255) / 256, 256, 0, stream>>>(ei, bitmap, E);
    k_degree_inv<<<NNODES / 256, 256, 0, stream>>>(bitmap, dinv);

    const int gemm_blocks = (NNODES / 16) * (DFEAT / 16) * 32 / 256; // 8192 waves / 8
    const int elem_blocks = NNODES * DFEAT / 256;

    const float* layer_in = x;
    for (int L = 0; L < 3; ++L) {
        k_linear_wmma<<<gemm_blocks, 256, 0, stream>>>(layer_in, Wm[L], bv[L], Z);
        k_aggregate<<<NNODES, 256, 0, stream>>>(Z, dinv, bitmap, H);
        k_bn_stats<<<NPART, DFEAT, 0, stream>>>(H, pSum, pSq);
        k_bn_finalize<<<1, DFEAT, 0, stream>>>(pSum, pSq, gm[L], bt[L], scale, shift);
        float* dst = (L == 2) ? (float*)d_out : H;
        k_bn_apply<<<elem_blocks, 256, 0, stream>>>(H, scale, shift, dst, (L < 2) ? 1 : 0);
        layer_in = dst;   // next layer reads BN output (H in-place for L=0,1)
    }
}